// IIRFilter_24945170055406
// MI455X (gfx1250) — compile-verified
//
#include <hip/hip_runtime.h>

typedef __attribute__((ext_vector_type(2))) float v2f;
typedef __attribute__((ext_vector_type(4))) float v4f;
typedef __attribute__((ext_vector_type(8))) float v8f;
typedef __attribute__((ext_vector_type(4))) unsigned int v4u;
typedef __attribute__((ext_vector_type(4))) int v4i;
typedef __attribute__((ext_vector_type(8))) int v8i;

namespace {
constexpr int kBdim   = 64;
constexpr int kCdim   = 2;
constexpr int kL      = 131072;
constexpr int kSeq    = kBdim * kCdim;     // 128 independent sequences
constexpr int kTile   = 256;               // 16 blocks of 16 samples = one WMMA tile
constexpr int kGroup  = 1024;              // output samples owned by one wave
constexpr int kWarm   = 768;               // warm-up samples (t^3 * 0.95^768 << fp32 eps)
constexpr int kGroups = kL / kGroup;       // 128
constexpr int kTilesOut  = kGroup / kTile; // 4
constexpr int kTilesWarm = kWarm / kTile;  // 3
constexpr int kWin    = kWarm + kGroup;    // 1792 floats staged per wave
// Per-sequence workspace layout (floats)
constexpr int kWStride = 704;
constexpr int kOffT = 0;    // T  : 16x16 forced-response matrix, row-major [t][k]
constexpr int kOffG = 256;  // G  : 16x8  state->output matrix,   row-major [t][i]
constexpr int kOffF = 384;  // Fp : 16x16 input->state matrix (rows 8..15 zero) [i][k]
constexpr int kOffA = 448;  // A16: 8x8   state hop A^16,         row-major [i][j]
}

static __device__ __forceinline__ v8f wmma_f32(v2f a, v2f b, v8f c) {
  // D = A(16x4,f32) x B(4x16,f32) + C(16x16,f32)
  return __builtin_amdgcn_wmma_f32_16x16x4_f32(false, a, false, b, (short)0, c,
                                               false, false);
}

// ---------------------------------------------------------------------------
// Phase 0: per-sequence block matrices by direct simulation of the exact
// reference recurrence (transposed direct-form II cascade).
//   lanes 0..15 : unit impulse at sample j, zero state  -> column j of T and F
//   lanes 16..23: unit initial state e_{j-16}, zero input -> column of G and A16
// ---------------------------------------------------------------------------
__global__ void __launch_bounds__(32) iir_build_mats(const float* __restrict__ Bs,
                                                     const float* __restrict__ As,
                                                     float* __restrict__ W) {
  const int seq = blockIdx.x;
  const int j   = threadIdx.x;

  float b0[4], b1[4], b2[4], a1[4], a2[4];
#pragma unroll
  for (int k = 0; k < 4; ++k) {
    const float* bp = Bs + (seq * 4 + k) * 3;
    const float* ap = As + (seq * 4 + k) * 3;
    const float inv = 1.0f / ap[0];
    b0[k] = bp[0] * inv; b1[k] = bp[1] * inv; b2[k] = bp[2] * inv;
    a1[k] = ap[1] * inv; a2[k] = ap[2] * inv;
  }

  float* Wm = W + seq * kWStride;

  const bool isImp = (j < 16);
  const int  sidx  = j - 16;
  if (!isImp && sidx >= 8) return;  // lanes 24..31 idle

  float s[8];
#pragma unroll
  for (int i = 0; i < 8; ++i) s[i] = 0.0f;
  if (!isImp) s[sidx] = 1.0f;

#pragma unroll
  for (int t = 0; t < 16; ++t) {
    float u = (isImp && (t == j)) ? 1.0f : 0.0f;
#pragma unroll
    for (int k = 0; k < 4; ++k) {
      const float yv  = fmaf(b0[k], u, s[2 * k]);
      const float s1n = fmaf(b1[k], u, s[2 * k + 1]) - a1[k] * yv;
      const float s2n = b2[k] * u - a2[k] * yv;
      s[2 * k] = s1n; s[2 * k + 1] = s2n;
      u = yv;
    }
    if (isImp) Wm[kOffT + t * 16 + j]   = u;  // T[t][j]
    else       Wm[kOffG + t * 8 + sidx] = u;  // G[t][sidx]
  }

  if (isImp) {
#pragma unroll
    for (int i = 0; i < 8; ++i)  Wm[kOffF + i * 16 + j] = s[i];   // F rows 0..7
#pragma unroll
    for (int i = 8; i < 16; ++i) Wm[kOffF + i * 16 + j] = 0.0f;   // zero pad
  } else {
#pragma unroll
    for (int i = 0; i < 8; ++i)  Wm[kOffA + i * 8 + sidx] = s[i]; // A16[i][sidx]
  }
}

// ---------------------------------------------------------------------------
// Phase 1: one wave per (sequence, 1024-sample group). Warm-start 768 samples
// early from zero state (group 0 is exact).
//   - TDM stages the wave's whole 1792-float input window into LDS
//     asynchronously while T/G/F/A16 constants load (A16 -> SGPRs).
//   - Per 256-sample tile: Sf = F*X (4 WMMA), Yf = T*X (4 WMMA, output tiles),
//     16-step 8x8 state-hop scan on VALU (co-executes with XDL),
//     Yf += G*S0 (2 WMMA), coalesced b128 stores.
// ---------------------------------------------------------------------------
__global__ void __launch_bounds__(32) iir_apply(const float* __restrict__ x,
                                                const float* __restrict__ W,
                                                float* __restrict__ y) {
  __shared__ float xlds[kWin];

  const int g    = blockIdx.x;
  const int seq  = blockIdx.y;
  const int lane = threadIdx.x;
  const int col  = lane & 15;   // N (block index) / M (A-matrix row)
  const int half = lane >> 4;   // 0: K even pair, 1: K odd pair
  const int off  = half * 2;

  const float* xs = x + (long long)seq * kL;
  float*       ys = y + (long long)seq * kL;

  int wstart, ntiles, firstOut;
  if (g == 0) { wstart = 0; ntiles = kTilesOut; firstOut = 0; }
  else {
    wstart = g * kGroup - kWarm;
    ntiles = kTilesOut + kTilesWarm;
    firstOut = kTilesWarm;
  }
  const unsigned winLen = (unsigned)(ntiles * kTile);  // elements (4B each)

  // ---- Tensor Data Mover: async DMA of the input window into LDS ----------
  // D# group0: count=1, lds_addr, global_addr[56:0], type=2 ("image")
  const unsigned long long gaddr = (unsigned long long)(uintptr_t)(xs + wstart);
  const unsigned laddr = (unsigned)(uintptr_t)(&xlds[0]);  // flat low 32 = LDS offset
  v4u g0 = { 1u,
             laddr,
             (unsigned)gaddr,
             (unsigned)((gaddr >> 32) & 0x01FFFFFFu) | (2u << 30) };
  // D# group1: data_size=4B; 2-D tensor winLen x 1; tile = winLen x 1
  v8i g1 = { (int)(2u << 16),                          // data_size=2 (4 bytes)
             (int)((winLen & 0xFFFFu) << 16),          // tensor_dim0[15:0]
             (int)((winLen >> 16) | (1u << 16)),       // tensor_dim0[31:16] | tensor_dim1[15:0]=1
             (int)(winLen << 16),                      // tensor_dim1[31:16]=0 | tile_dim0
             1,                                        // tile_dim1=1, tile_dim2=0
             (int)winLen,                              // tensor_dim0_stride[31:0]
             (int)((winLen & 0xFFFFu) << 16),          // stride0[47:32]=0 | stride1[15:0]
             (int)(winLen >> 16) };                    // stride1[47:16]
  v4i gz4 = { 0, 0, 0, 0 };                            // groups 2/3 unused (<=2-D)
  v8i gz8 = { 0, 0, 0, 0, 0, 0, 0, 0 };                // trailing group (6-arg form)
  __builtin_amdgcn_tensor_load_to_lds(g0, g1, gz4, gz4, gz8, 0);

  // ---- constant loads overlap the DMA -------------------------------------
  const float* Wm = W + seq * kWStride;
  v2f Ta[4], Fa[4], Ga[2];
#pragma unroll
  for (int c = 0; c < 4; ++c) {
    Ta[c] = *(const v2f*)(Wm + kOffT + col * 16 + 4 * c + off);
    Fa[c] = *(const v2f*)(Wm + kOffF + col * 16 + 4 * c + off);
  }
#pragma unroll
  for (int c = 0; c < 2; ++c)
    Ga[c] = *(const v2f*)(Wm + kOffG + col * 8 + 4 * c + off);

  float Am[64];  // A^16, wave-uniform -> scalar loads / SGPRs
#pragma unroll
  for (int i = 0; i < 64; ++i) Am[i] = Wm[kOffA + i];

  float s[8];
#pragma unroll
  for (int i = 0; i < 8; ++i) s[i] = 0.0f;

  __builtin_amdgcn_s_wait_tensorcnt(0);  // window resident in LDS

  for (int tile = 0; tile < ntiles; ++tile) {
    const int  tstart = wstart + tile * kTile;
    const bool isOut  = (tile >= firstOut);

    // B-matrix X from LDS: lane supplies X[4c+off][col], X[4c+off+1][col]
    const float* xt = &xlds[tile * kTile + col * 16 + off];
    v2f Xb[4];
#pragma unroll
    for (int c = 0; c < 4; ++c) Xb[c] = *(const v2f*)(xt + 4 * c);

    // Sf(8x16) = F * X : per-block state injections
    v8f sf = {0.f, 0.f, 0.f, 0.f, 0.f, 0.f, 0.f, 0.f};
#pragma unroll
    for (int c = 0; c < 4; ++c) sf = wmma_f32(Fa[c], Xb[c], sf);

    // Yf(16x16) = T * X : forced response (output tiles only; wave-uniform)
    v8f yacc = {0.f, 0.f, 0.f, 0.f, 0.f, 0.f, 0.f, 0.f};
    if (isOut) {
#pragma unroll
      for (int c = 0; c < 4; ++c) yacc = wmma_f32(Ta[c], Xb[c], yacc);
    }

    // Sequential 8-dim hop across the 16 blocks; lane snapshots the state at
    // the start of its own block (j == col) for the G*S0 correction.
    float s0m[8];
#pragma unroll
    for (int i = 0; i < 8; ++i) s0m[i] = 0.0f;
#pragma unroll
    for (int j = 0; j < 16; ++j) {
      if (isOut) {
        const bool mine = (j == col);
#pragma unroll
        for (int i = 0; i < 8; ++i) s0m[i] = mine ? s[i] : s0m[i];
      }
      float ns[8];
#pragma unroll
      for (int i = 0; i < 8; ++i) {
        float acc = __shfl(sf[i], j, 32);   // Sf[i][j] lives in lane j, vgpr i
#pragma unroll
        for (int jj = 0; jj < 8; ++jj) acc = fmaf(Am[i * 8 + jj], s[jj], acc);
        ns[i] = acc;
      }
#pragma unroll
      for (int i = 0; i < 8; ++i) s[i] = ns[i];
    }

    if (isOut) {
      // Y += G(16x8) * S0(8x16); B operand assembled from the lane snapshot
      v2f B0 = { half ? s0m[2] : s0m[0], half ? s0m[3] : s0m[1] };
      v2f B1 = { half ? s0m[6] : s0m[4], half ? s0m[7] : s0m[5] };
      yacc = wmma_f32(Ga[0], B0, yacc);
      yacc = wmma_f32(Ga[1], B1, yacc);

      // D layout: vgpr v -> sample t = v + 8*half of block `col`
      float* yp = ys + tstart + col * 16 + half * 8;
      v4f o0 = { yacc[0], yacc[1], yacc[2], yacc[3] };
      v4f o1 = { yacc[4], yacc[5], yacc[6], yacc[7] };
      *(v4f*)(yp)     = o0;
      *(v4f*)(yp + 4) = o1;
    }
  }
}

extern "C" void kernel_launch(void* const* d_in, const int* in_sizes, int n_in,
                              void* d_out, int out_size, void* d_ws, size_t ws_size,
                              hipStream_t stream) {
  (void)in_sizes; (void)n_in; (void)out_size; (void)ws_size;
  const float* x  = (const float*)d_in[0];
  const float* Bs = (const float*)d_in[1];
  const float* As = (const float*)d_in[2];
  float* y = (float*)d_out;
  float* W = (float*)d_ws;   // 128 * 704 floats = 352 KB

  iir_build_mats<<<dim3(kSeq), dim3(32), 0, stream>>>(Bs, As, W);
  iir_apply<<<dim3(kGroups, kSeq), dim3(32), 0, stream>>>(x, W, y);
}